// SelfAttention2d_5677946765932
// MI455X (gfx1250) — compile-verified
//
#include <hip/hip_runtime.h>
#include <hip/hip_bf16.h>
#include <math.h>

typedef __attribute__((ext_vector_type(2))) float v2f;
typedef __attribute__((ext_vector_type(8))) float v8f;

// ---------------------------------------------------------------------------
// Kernel 1: GroupNorm.  x[4,512,32,32], 16 groups of 32 channels per batch.
// One block per (n, group): 32 ch * 1024 px = 32768 elements.
// ---------------------------------------------------------------------------
__global__ __launch_bounds__(256) void gn_kernel(
    const float* __restrict__ X, const float* __restrict__ gw,
    const float* __restrict__ gb, float* __restrict__ Xn) {
  __shared__ float ssum[256];
  __shared__ float ssq[256];
  const int tid = threadIdx.x;
  const int n = blockIdx.x >> 4;
  const int g = blockIdx.x & 15;
  const size_t base = ((size_t)n * 512 + (size_t)g * 32) * 1024;
  float s = 0.f, q = 0.f;
  for (int i = tid; i < 32768; i += 256) {
    float v = X[base + i];
    s += v;
    q += v * v;
  }
  ssum[tid] = s;
  ssq[tid] = q;
  __syncthreads();
  for (int off = 128; off > 0; off >>= 1) {
    if (tid < off) {
      ssum[tid] += ssum[tid + off];
      ssq[tid] += ssq[tid + off];
    }
    __syncthreads();
  }
  const float mean = ssum[0] * (1.0f / 32768.0f);
  const float var = ssq[0] * (1.0f / 32768.0f) - mean * mean;
  const float rstd = rsqrtf(var + 1e-5f);
  for (int i = tid; i < 32768; i += 256) {
    int c = g * 32 + (i >> 10);
    float v = (X[base + i] - mean) * rstd;
    Xn[base + i] = v * gw[c] + gb[c];
  }
}

// ---------------------------------------------------------------------------
// Shared GEMM tile loop (2x2 register blocking per wave).
// Block = 8 waves, covers 32 rows x 256 cols of C.  K = 512 in KC chunks.
// A fragments reused across 2 col-tiles, B fragments across 2 row-tiles:
// 4 WMMAs per 4 LDS fragment pairs (half the LDS bytes per WMMA of a 1x1
// scheme) -- LDS bandwidth is the limiter when feeding v_wmma_f32_16x16x4.
// ---------------------------------------------------------------------------
#define KC 64

// ---------------------------------------------------------------------------
// Kernel 2: QKV GEMM.  qkv[n][d][p] = sum_c W[d][c]*xn[n][c][p] + b[d]
// W: [1536][512], xn: [4][512][1024], out: QKV [4][3][512][1024] channel-major
// ---------------------------------------------------------------------------
__global__ __launch_bounds__(256) void qkv_gemm_kernel(
    const float* __restrict__ W, const float* __restrict__ bias,
    const float* __restrict__ Xn, float* __restrict__ QKV) {
  __shared__ float As[32][KC + 1];
  __shared__ float Bs[KC][256 + 1];
  const int tid = threadIdx.x;
  const int lane = tid & 31;
  const int wave = tid >> 5;
  const int hi = lane >> 4;   // half index within wave32
  const int col = lane & 15;  // A-row / B-col per WMMA layout
  const int row0 = blockIdx.x * 32;   // 0..1535 step 32
  const int nb = blockIdx.y * 256;    // pixel base
  const int n = blockIdx.z;
  const float* Xb = Xn + (size_t)n * 512 * 1024 + nb;
  v8f acc00 = {}, acc01 = {}, acc10 = {}, acc11 = {};
  for (int k0 = 0; k0 < 512; k0 += KC) {
    __syncthreads();
    for (int i = tid; i < 32 * KC; i += 256) {
      int r = i / KC, c = i % KC;
      As[r][c] = W[(size_t)(row0 + r) * 512 + k0 + c];
    }
    for (int i = tid; i < KC * 256; i += 256) {
      int r = i >> 8, c = i & 255;
      Bs[r][c] = Xb[(size_t)(k0 + r) * 1024 + c];
    }
    __syncthreads();
    if (k0 + KC < 512) {  // prefetch next chunk (global_prefetch_b8)
      int r = tid >> 3;   // 32 rows of A
      __builtin_prefetch(&W[(size_t)(row0 + r) * 512 + k0 + KC], 0, 1);
      __builtin_prefetch(&Xb[(size_t)(k0 + KC + (tid >> 2)) * 1024], 0, 1);
    }
    const int wc0 = wave * 16 + col;
    const int wc1 = 128 + wave * 16 + col;
#pragma unroll
    for (int kk = 0; kk < KC; kk += 4) {
      v2f a0, a1, b0, b1;
      a0.x = As[col][kk + 2 * hi];
      a0.y = As[col][kk + 2 * hi + 1];
      a1.x = As[16 + col][kk + 2 * hi];
      a1.y = As[16 + col][kk + 2 * hi + 1];
      b0.x = Bs[kk + 2 * hi][wc0];
      b0.y = Bs[kk + 2 * hi + 1][wc0];
      b1.x = Bs[kk + 2 * hi][wc1];
      b1.y = Bs[kk + 2 * hi + 1][wc1];
      acc00 = __builtin_amdgcn_wmma_f32_16x16x4_f32(false, a0, false, b0,
                                                    (short)0, acc00, false, false);
      acc01 = __builtin_amdgcn_wmma_f32_16x16x4_f32(false, a0, false, b1,
                                                    (short)0, acc01, false, false);
      acc10 = __builtin_amdgcn_wmma_f32_16x16x4_f32(false, a1, false, b0,
                                                    (short)0, acc10, false, false);
      acc11 = __builtin_amdgcn_wmma_f32_16x16x4_f32(false, a1, false, b1,
                                                    (short)0, acc11, false, false);
    }
  }
#pragma unroll
  for (int rh = 0; rh < 2; rh++) {
#pragma unroll
    for (int ch = 0; ch < 2; ch++) {
      const v8f& acc = rh ? (ch ? acc11 : acc10) : (ch ? acc01 : acc00);
#pragma unroll
      for (int j = 0; j < 8; j++) {
        int d = row0 + 16 * rh + j + 8 * hi;  // 0..1535
        int p = nb + 128 * ch + wave * 16 + col;
        int s3 = d >> 9;  // 0..2 (q/k/v)
        int cc = d & 511;
        QKV[(((size_t)n * 3 + s3) * 512 + cc) * 1024 + p] = acc[j] + bias[d];
      }
    }
  }
}

// ---------------------------------------------------------------------------
// Kernel 3: flash attention.  64 heads, hd=8, seq=1024.
// Block = (8 query-strips, 64 heads, 4 batches); wave = one 16-query tile.
// Streams 16-key tiles: S=QK^T (2 WMMA) -> online softmax (shfl_xor within
// 16-lane halves) -> P via LDS transpose -> O += P@V (4 WMMA, V padded N=16).
// Never materializes the 4*64*1024*1024 (1 GB) score tensor.
// ---------------------------------------------------------------------------
__global__ __launch_bounds__(256) void attn_kernel(
    const float* __restrict__ QKV, float* __restrict__ Y) {
  __shared__ float Kt[16][9];      // [key][dim]
  __shared__ float Vt[16][17];     // [key][dim padded to 16, upper 8 zero]
  __shared__ float Pt[8][16][17];  // per-wave P transpose staging
  const int tid = threadIdx.x;
  const int lane = tid & 31;
  const int wave = tid >> 5;
  const int hi = lane >> 4;
  const int col = lane & 15;
  const int n = blockIdx.z;
  const int head = blockIdx.y;
  const int q0 = blockIdx.x * 128 + wave * 16;
  const size_t nbase = (size_t)n * 3 * 512 * 1024;
  const float* Qp = QKV + nbase + (size_t)(head * 8) * 1024;
  const float* Kp = QKV + nbase + (size_t)(512 + head * 8) * 1024;
  const float* Vp = QKV + nbase + (size_t)(1024 + head * 8) * 1024;

  // zero V padding lanes (dims 8..15 never overwritten below)
  for (int i = tid; i < 16 * 17; i += 256) ((float*)Vt)[i] = 0.f;

  // Q tile in A layout: K-dim = head dim (8) -> 2 WMMA steps
  v2f qa[2];
  const int qrow = q0 + col;
#pragma unroll
  for (int kt = 0; kt < 2; kt++) {
    int kd = 4 * kt + 2 * hi;
    qa[kt].x = Qp[(size_t)kd * 1024 + qrow];
    qa[kt].y = Qp[(size_t)(kd + 1) * 1024 + qrow];
  }

  float m[8], lsum[8];
#pragma unroll
  for (int j = 0; j < 8; j++) {
    m[j] = -1e30f;
    lsum[j] = 0.f;
  }
  v8f o = {};
  const float scale = 0.35355339059327373f;  // 1/sqrt(8)

  for (int t = 0; t < 64; t++) {
    __syncthreads();
    for (int i = tid; i < 128; i += 256) {  // 16 keys x 8 dims
      int key = i >> 3, dim = i & 7;
      int p = t * 16 + key;
      Kt[key][dim] = Kp[(size_t)dim * 1024 + p];
      Vt[key][dim] = Vp[(size_t)dim * 1024 + p];
    }
    __syncthreads();

    // S = Q K^T  (B[k=dim][n=key] = Kt[key][dim])
    v8f s = {};
#pragma unroll
    for (int kt = 0; kt < 2; kt++) {
      int kd = 4 * kt + 2 * hi;
      v2f b;
      b.x = Kt[col][kd];
      b.y = Kt[col][kd + 1];
      s = __builtin_amdgcn_wmma_f32_16x16x4_f32(false, qa[kt], false, b,
                                                (short)0, s, false, false);
    }

    // online softmax: row j+8*hi lives in VGPR j across this 16-lane half
    float pv[8];
#pragma unroll
    for (int j = 0; j < 8; j++) {
      float sj = s[j] * scale;
      float r = sj;
      r = fmaxf(r, __shfl_xor(r, 1));
      r = fmaxf(r, __shfl_xor(r, 2));
      r = fmaxf(r, __shfl_xor(r, 4));
      r = fmaxf(r, __shfl_xor(r, 8));
      float mn = fmaxf(m[j], r);
      float f = __expf(m[j] - mn);
      float p = __expf(sj - mn);
      float rs = p;
      rs += __shfl_xor(rs, 1);
      rs += __shfl_xor(rs, 2);
      rs += __shfl_xor(rs, 4);
      rs += __shfl_xor(rs, 8);
      lsum[j] = lsum[j] * f + rs;
      m[j] = mn;
      o[j] *= f;
      pv[j] = p;
    }

    // transpose P (D layout -> A layout) through wave-private LDS tile
#pragma unroll
    for (int j = 0; j < 8; j++) Pt[wave][j + 8 * hi][col] = pv[j];
    __syncthreads();

    // O += P @ V   (K = 16 keys -> 4 WMMA steps; V cols 8..15 are zero)
#pragma unroll
    for (int kt = 0; kt < 4; kt++) {
      int kk = 4 * kt + 2 * hi;
      v2f a, b;
      a.x = Pt[wave][col][kk];
      a.y = Pt[wave][col][kk + 1];
      b.x = Vt[kk][col];
      b.y = Vt[kk + 1][col];
      o = __builtin_amdgcn_wmma_f32_16x16x4_f32(false, a, false, b, (short)0,
                                                o, false, false);
    }
  }

  // normalize and store y channel-major: Y[n][head*8+dim][query]
  if (col < 8) {
#pragma unroll
    for (int j = 0; j < 8; j++) {
      int qr = q0 + j + 8 * hi;
      Y[((size_t)n * 512 + head * 8 + col) * 1024 + qr] = o[j] / lsum[j];
    }
  }
}

// ---------------------------------------------------------------------------
// Kernel 4: out-proj GEMM + bias + residual with xn (2x2 blocked like K2).
// out[n][d][p] = xn[n][d][p] + sum_c W[d][c]*y[n][c][p] + b[d]
// ---------------------------------------------------------------------------
__global__ __launch_bounds__(256) void out_gemm_kernel(
    const float* __restrict__ W, const float* __restrict__ bias,
    const float* __restrict__ Yb, const float* __restrict__ Xn,
    float* __restrict__ Out) {
  __shared__ float As[32][KC + 1];
  __shared__ float Bs[KC][256 + 1];
  const int tid = threadIdx.x;
  const int lane = tid & 31;
  const int wave = tid >> 5;
  const int hi = lane >> 4;
  const int col = lane & 15;
  const int row0 = blockIdx.x * 32;  // 0..511 step 32
  const int nb = blockIdx.y * 256;
  const int n = blockIdx.z;
  const float* Yn = Yb + (size_t)n * 512 * 1024 + nb;
  v8f acc00 = {}, acc01 = {}, acc10 = {}, acc11 = {};
  for (int k0 = 0; k0 < 512; k0 += KC) {
    __syncthreads();
    for (int i = tid; i < 32 * KC; i += 256) {
      int r = i / KC, c = i % KC;
      As[r][c] = W[(size_t)(row0 + r) * 512 + k0 + c];
    }
    for (int i = tid; i < KC * 256; i += 256) {
      int r = i >> 8, c = i & 255;
      Bs[r][c] = Yn[(size_t)(k0 + r) * 1024 + c];
    }
    __syncthreads();
    if (k0 + KC < 512) {
      int r = tid >> 3;
      __builtin_prefetch(&W[(size_t)(row0 + r) * 512 + k0 + KC], 0, 1);
      __builtin_prefetch(&Yn[(size_t)(k0 + KC + (tid >> 2)) * 1024], 0, 1);
    }
    const int wc0 = wave * 16 + col;
    const int wc1 = 128 + wave * 16 + col;
#pragma unroll
    for (int kk = 0; kk < KC; kk += 4) {
      v2f a0, a1, b0, b1;
      a0.x = As[col][kk + 2 * hi];
      a0.y = As[col][kk + 2 * hi + 1];
      a1.x = As[16 + col][kk + 2 * hi];
      a1.y = As[16 + col][kk + 2 * hi + 1];
      b0.x = Bs[kk + 2 * hi][wc0];
      b0.y = Bs[kk + 2 * hi + 1][wc0];
      b1.x = Bs[kk + 2 * hi][wc1];
      b1.y = Bs[kk + 2 * hi + 1][wc1];
      acc00 = __builtin_amdgcn_wmma_f32_16x16x4_f32(false, a0, false, b0,
                                                    (short)0, acc00, false, false);
      acc01 = __builtin_amdgcn_wmma_f32_16x16x4_f32(false, a0, false, b1,
                                                    (short)0, acc01, false, false);
      acc10 = __builtin_amdgcn_wmma_f32_16x16x4_f32(false, a1, false, b0,
                                                    (short)0, acc10, false, false);
      acc11 = __builtin_amdgcn_wmma_f32_16x16x4_f32(false, a1, false, b1,
                                                    (short)0, acc11, false, false);
    }
  }
#pragma unroll
  for (int rh = 0; rh < 2; rh++) {
#pragma unroll
    for (int ch = 0; ch < 2; ch++) {
      const v8f& acc = rh ? (ch ? acc11 : acc10) : (ch ? acc01 : acc00);
#pragma unroll
      for (int j = 0; j < 8; j++) {
        int d = row0 + 16 * rh + j + 8 * hi;
        int p = nb + 128 * ch + wave * 16 + col;
        size_t idx = ((size_t)n * 512 + d) * 1024 + p;
        Out[idx] = Xn[idx] + acc[j] + bias[d];
      }
    }
  }
}

// ---------------------------------------------------------------------------
extern "C" void kernel_launch(void* const* d_in, const int* in_sizes, int n_in,
                              void* d_out, int out_size, void* d_ws,
                              size_t ws_size, hipStream_t stream) {
  const float* x = (const float*)d_in[0];
  const float* gn_w = (const float*)d_in[1];
  const float* gn_b = (const float*)d_in[2];
  const float* qkv_w = (const float*)d_in[3];
  const float* qkv_b = (const float*)d_in[4];
  const float* out_w = (const float*)d_in[5];
  const float* out_b = (const float*)d_in[6];
  float* out = (float*)d_out;

  float* ws = (float*)d_ws;
  float* xn = ws;                      // 4*512*1024   = 2,097,152 f
  float* qkv = ws + 2097152;           // 4*3*512*1024 = 6,291,456 f
  float* yb = ws + 2097152 + 6291456;  // 4*512*1024   = 2,097,152 f

  gn_kernel<<<64, 256, 0, stream>>>(x, gn_w, gn_b, xn);
  qkv_gemm_kernel<<<dim3(48, 4, 4), 256, 0, stream>>>(qkv_w, qkv_b, xn, qkv);
  attn_kernel<<<dim3(8, 64, 4), 256, 0, stream>>>(qkv, yb);
  out_gemm_kernel<<<dim3(16, 4, 4), 256, 0, stream>>>(out_w, out_b, yb, xn,
                                                      out);
}